// PointNetModule_25400436588648
// MI455X (gfx1250) — compile-verified
//
#include <hip/hip_runtime.h>
#include <hip/hip_bf16.h>

typedef __attribute__((ext_vector_type(2))) float v2f;
typedef __attribute__((ext_vector_type(8))) float v8f;

#define GN_EPS 1e-5f
#define WPB 4                 // waves per block in tile kernels (wave32)
#define TILE_THREADS (WPB * 32)

// ---------------------------------------------------------------------------
// Workspace layout (bytes, from d_ws):
//   segmax : V*64 f32   (per-vertex per-channel max, init 0; values >= 0)
//   argidx : V*64 i32   (min row index achieving max, init N)
//   counts : V    i32   (points per vertex, init 0), padded to 256B
//   stats  : 256  f32   [0..2]=sum x, [3..8]=sum {xx,xy,xz,yy,yz,zz},
//                        [16..79]=sum y1[c], [80..143]=sum y1[c]^2
//   P0     : 64*4 f32   fused layer0: {a0,a1,a2,bias} per channel (GN0 folded)
//   P1     : 64*2 f32   layer1 GN affine {scale, bias} per channel
// ---------------------------------------------------------------------------

__global__ __launch_bounds__(256) void k_init(float* segmax, int* argidx,
                                              int* counts, float* stats,
                                              int V, int N) {
  int t = blockIdx.x * blockDim.x + threadIdx.x;
  int tot = V * 64;
  if (t < tot) { segmax[t] = 0.0f; argidx[t] = N; }
  if (t < V) counts[t] = 0;
  if (t < 256) stats[t] = 0.0f;
}

// Pass 1: moments of raw 3-d input + per-vertex counts.
__global__ __launch_bounds__(256) void k_pass1(const float4* __restrict__ dist,
                                               const int* __restrict__ indices,
                                               int* counts, float* stats,
                                               int N, int V) {
  float s[9] = {0.f, 0.f, 0.f, 0.f, 0.f, 0.f, 0.f, 0.f, 0.f};
  for (int i = blockIdx.x * blockDim.x + threadIdx.x; i < N;
       i += gridDim.x * blockDim.x) {
    float4 d = dist[i];
    s[0] += d.x; s[1] += d.y; s[2] += d.z;
    s[3] += d.x * d.x; s[4] += d.x * d.y; s[5] += d.x * d.z;
    s[6] += d.y * d.y; s[7] += d.y * d.z; s[8] += d.z * d.z;
    int id = indices[i];
    id = min(max(id, 0), V - 1);
    atomicAdd(&counts[id], 1);
  }
#pragma unroll
  for (int k = 0; k < 9; ++k)
    for (int o = 16; o > 0; o >>= 1) s[k] += __shfl_xor(s[k], o);
  __shared__ float red[9];
  if (threadIdx.x < 9) red[threadIdx.x] = 0.f;
  __syncthreads();
  if ((threadIdx.x & 31) == 0)
    for (int k = 0; k < 9; ++k) atomicAdd(&red[k], s[k]);
  __syncthreads();
  if (threadIdx.x < 9) atomicAdd(&stats[threadIdx.x], red[threadIdx.x]);
}

// Layer-0 GN stats analytically from input moments (y0 = x @ W0^T is linear).
__global__ __launch_bounds__(64) void k_params0(const float* __restrict__ stats,
                                                const float* __restrict__ W0,
                                                const float* __restrict__ g0,
                                                const float* __restrict__ beta0,
                                                float* P0, int N) {
  int c = threadIdx.x;
  if (c >= 64) return;
  float invN = 1.0f / (float)N;
  float mx = stats[0] * invN, my = stats[1] * invN, mz = stats[2] * invN;
  float Sxx = stats[3] * invN, Sxy = stats[4] * invN, Sxz = stats[5] * invN;
  float Syy = stats[6] * invN, Syz = stats[7] * invN, Szz = stats[8] * invN;
  int g = c >> 1;
  float gm = 0.f, gq = 0.f;
#pragma unroll
  for (int j = 0; j < 2; ++j) {
    int cc = 2 * g + j;
    float w0 = W0[cc * 3 + 0], w1 = W0[cc * 3 + 1], w2 = W0[cc * 3 + 2];
    float m  = w0 * mx + w1 * my + w2 * mz;
    float e2 = w0 * w0 * Sxx + w1 * w1 * Syy + w2 * w2 * Szz +
               2.f * (w0 * w1 * Sxy + w0 * w2 * Sxz + w1 * w2 * Syz);
    gm += 0.5f * m;
    gq += 0.5f * e2;
  }
  float var = gq - gm * gm;
  float sc = g0[c] * rsqrtf(var + GN_EPS);
  P0[c * 4 + 0] = sc * W0[c * 3 + 0];
  P0[c * 4 + 1] = sc * W0[c * 3 + 1];
  P0[c * 4 + 2] = sc * W0[c * 3 + 2];
  P0[c * 4 + 3] = beta0[c] - gm * sc;
}

// Pass 2: sums of y1 = relu(gn0(x@W0^T)) @ W1^T via fp32 WMMA 16x16x4.
__global__ __launch_bounds__(TILE_THREADS) void k_stats1(
    const float4* __restrict__ dist, const float* __restrict__ P0g,
    const float* __restrict__ W1, float* stats, int N) {
  __shared__ float W1T[4096];       // W1T[k*64+c] = W1[c*64+k]
  __shared__ float4 P0s[64];
  __shared__ float Xstage[WPB][16 * 68];
  for (int j = threadIdx.x; j < 4096; j += blockDim.x) {
    int k = j >> 6, c = j & 63;
    W1T[j] = W1[c * 64 + k];
  }
  for (int j = threadIdx.x; j < 64; j += blockDim.x)
    P0s[j] = ((const float4*)P0g)[j];
  __syncthreads();

  const int lane = threadIdx.x & 31;
  const int wave = threadIdx.x >> 5;
  const int hi = lane >> 4;       // half-wave select (K / M-row split)
  const int lr = lane & 15;
  float* Xs = Xstage[wave];

  const int numTiles = (N + 15) >> 4;
  const int waveId = blockIdx.x * WPB + wave;
  const int waveStride = gridDim.x * WPB;

  float sum[4] = {0.f, 0.f, 0.f, 0.f};
  float ssq[4] = {0.f, 0.f, 0.f, 0.f};

  for (int t = waveId; t < numTiles; t += waveStride) {
    const int base = t << 4;
    // opaque zero: redefined every iteration -> LDS table loads cannot be
    // hoisted (and spilled) out of the tile loop by LICM.
    int zero = 0;
    asm volatile("" : "+v"(zero));

    const bool rv = (base + lr) < N;              // row validity (tail only)
    const float4 d = dist[rv ? base + lr : N - 1];
#pragma unroll
    for (int cp = 0; cp < 32; ++cp) {             // X1 = relu(A0 . x + c0)
      const int c = 2 * cp + hi;
      const float4 p = P0s[c + zero];
      float v = fmaf(p.x, d.x, fmaf(p.y, d.y, fmaf(p.z, d.z, p.w)));
      v = fmaxf(v, 0.0f);
      Xs[lr * 68 + c] = rv ? v : 0.0f;            // zero rows past N: y1 = 0
    }
    v2f a[16];
#pragma unroll
    for (int s = 0; s < 16; ++s) {                // A-fragments (16x4 f32)
      const int kk = 4 * s + 2 * hi;
      a[s][0] = Xs[lr * 68 + kk];
      a[s][1] = Xs[lr * 68 + kk + 1];
    }
#pragma unroll
    for (int cb = 0; cb < 4; ++cb) {
      v8f acc = {0.f, 0.f, 0.f, 0.f, 0.f, 0.f, 0.f, 0.f};
#pragma unroll
      for (int s = 0; s < 16; ++s) {
        const int kk = 4 * s + 2 * hi;
        v2f b;
        b[0] = W1T[kk * 64 + cb * 16 + lr + zero];
        b[1] = W1T[(kk + 1) * 64 + cb * 16 + lr + zero];
        acc = __builtin_amdgcn_wmma_f32_16x16x4_f32(false, a[s], false, b,
                                                    (short)0, acc, false, false);
      }
#pragma unroll
      for (int v = 0; v < 8; ++v) {               // unconditional: pad rows are 0
        const float y = acc[v];
        sum[cb] += y;
        ssq[cb] += y * y;
      }
    }
  }
#pragma unroll
  for (int cb = 0; cb < 4; ++cb) {
    float s = sum[cb] + __shfl_xor(sum[cb], 16);
    float q = ssq[cb] + __shfl_xor(ssq[cb], 16);
    if (lane < 16) {
      atomicAdd(&stats[16 + cb * 16 + lane], s);
      atomicAdd(&stats[80 + cb * 16 + lane], q);
    }
  }
}

__global__ __launch_bounds__(64) void k_params1(const float* __restrict__ stats,
                                                const float* __restrict__ g1,
                                                const float* __restrict__ beta1,
                                                float* P1, int N) {
  int c = threadIdx.x;
  if (c >= 64) return;
  float invN = 1.0f / (float)N;
  float m = stats[16 + c] * invN;
  float q = stats[80 + c] * invN;
  float mp = __shfl_xor(m, 1);           // group partner channel (c ^ 1)
  float qp = __shfl_xor(q, 1);
  float gm = 0.5f * (m + mp);
  float gq = 0.5f * (q + qp);
  float var = gq - gm * gm;
  float sc = g1[c] * rsqrtf(var + GN_EPS);
  P1[2 * c + 0] = sc;
  P1[2 * c + 1] = beta1[c] - gm * sc;
}

// Pass 3: full MLP forward per 16-row tile (two 64x64 WMMA GEMMs).
// mode 0: atomicMax into segmax.  mode 1: bit-identical recompute -> argmin idx.
__global__ __launch_bounds__(TILE_THREADS) void k_pass3(
    const float4* __restrict__ dist, const int* __restrict__ indices,
    const float* __restrict__ P0g, const float* __restrict__ W1,
    const float* __restrict__ W2, const float* __restrict__ P1g,
    const float* __restrict__ b2, float* segmax, int* argidx,
    int N, int V, int mode) {
  __shared__ float W1T[4096];
  __shared__ float W2T[4096];
  __shared__ float4 P0s[64];
  __shared__ float Xstage[WPB][16 * 68];
  for (int j = threadIdx.x; j < 4096; j += blockDim.x) {
    int k = j >> 6, c = j & 63;
    W1T[j] = W1[c * 64 + k];
    W2T[j] = W2[c * 64 + k];
  }
  for (int j = threadIdx.x; j < 64; j += blockDim.x)
    P0s[j] = ((const float4*)P0g)[j];
  __syncthreads();

  const int lane = threadIdx.x & 31;
  const int wave = threadIdx.x >> 5;
  const int hi = lane >> 4;
  const int lr = lane & 15;
  float* Xs = Xstage[wave];

  float s1v[4], c1v[4], b2v[4];          // per-lane channel params (c = cb*16+lr)
#pragma unroll
  for (int cb = 0; cb < 4; ++cb) {
    int c = cb * 16 + lr;
    s1v[cb] = P1g[2 * c];
    c1v[cb] = P1g[2 * c + 1];
    b2v[cb] = b2[c];
  }

  const int numTiles = (N + 15) >> 4;
  const int waveId = blockIdx.x * WPB + wave;
  const int waveStride = gridDim.x * WPB;

  for (int t = waveId; t < numTiles; t += waveStride) {
    const int base = t << 4;
    const int myRow0 = base + 8 * hi;    // first of this lane's 8 D-rows
    int zero = 0;
    asm volatile("" : "+v"(zero));       // defeat LICM of LDS table loads

    int vid[8];
#pragma unroll
    for (int v = 0; v < 8; ++v) {
      int ii = indices[min(myRow0 + v, N - 1)];
      vid[v] = min(max(ii, 0), V - 1);
    }
    // layer0: X1 tile into LDS
    const bool rv = (base + lr) < N;
    const float4 d = dist[rv ? base + lr : N - 1];
#pragma unroll
    for (int cp = 0; cp < 32; ++cp) {
      const int c = 2 * cp + hi;
      const float4 p = P0s[c + zero];
      float v = fmaf(p.x, d.x, fmaf(p.y, d.y, fmaf(p.z, d.z, p.w)));
      v = fmaxf(v, 0.0f);
      Xs[lr * 68 + c] = rv ? v : 0.0f;
    }
    v2f a[16];
#pragma unroll
    for (int s = 0; s < 16; ++s) {
      const int kk = 4 * s + 2 * hi;
      a[s][0] = Xs[lr * 68 + kk];
      a[s][1] = Xs[lr * 68 + kk + 1];
    }
    // layer1 GEMM + GN1 affine + relu -> X2 back into LDS stage
#pragma unroll
    for (int cb = 0; cb < 4; ++cb) {
      v8f acc = {0.f, 0.f, 0.f, 0.f, 0.f, 0.f, 0.f, 0.f};
#pragma unroll
      for (int s = 0; s < 16; ++s) {
        const int kk = 4 * s + 2 * hi;
        v2f b;
        b[0] = W1T[kk * 64 + cb * 16 + lr + zero];
        b[1] = W1T[(kk + 1) * 64 + cb * 16 + lr + zero];
        acc = __builtin_amdgcn_wmma_f32_16x16x4_f32(false, a[s], false, b,
                                                    (short)0, acc, false, false);
      }
#pragma unroll
      for (int v = 0; v < 8; ++v) {
        const int m = v + 8 * hi;
        Xs[m * 68 + cb * 16 + lr] =
            fmaxf(fmaf(s1v[cb], acc[v], c1v[cb]), 0.0f);
      }
    }
    // A-fragments of X2
#pragma unroll
    for (int s = 0; s < 16; ++s) {
      const int kk = 4 * s + 2 * hi;
      a[s][0] = Xs[lr * 68 + kk];
      a[s][1] = Xs[lr * 68 + kk + 1];
    }
    // layer2 GEMM + bias + relu -> segment reductions
#pragma unroll
    for (int cb = 0; cb < 4; ++cb) {
      v8f acc = {0.f, 0.f, 0.f, 0.f, 0.f, 0.f, 0.f, 0.f};
#pragma unroll
      for (int s = 0; s < 16; ++s) {
        const int kk = 4 * s + 2 * hi;
        v2f b;
        b[0] = W2T[kk * 64 + cb * 16 + lr + zero];
        b[1] = W2T[(kk + 1) * 64 + cb * 16 + lr + zero];
        acc = __builtin_amdgcn_wmma_f32_16x16x4_f32(false, a[s], false, b,
                                                    (short)0, acc, false, false);
      }
      const int c = cb * 16 + lr;
#pragma unroll
      for (int v = 0; v < 8; ++v) {
        const int r = myRow0 + v;
        if (r < N) {
          // +0.0f canonicalizes -0 -> +0 so uint atomicMax is order-correct
          float y = fmaxf(acc[v] + b2v[cb], 0.0f) + 0.0f;
          size_t o = (size_t)vid[v] * 64 + c;
          if (mode == 0) {
            atomicMax((unsigned int*)&segmax[o], __float_as_uint(y));
          } else {
            if (y == segmax[o]) atomicMin(&argidx[o], r);
          }
        }
      }
    }
  }
}

__global__ __launch_bounds__(256) void k_finalize(
    const float4* __restrict__ dist, const float* __restrict__ segmax,
    const int* __restrict__ argidx, const int* __restrict__ counts,
    float* out, int V, int N) {
  int t = blockIdx.x * blockDim.x + threadIdx.x;
  if (t >= V * 64) return;
  int v = t >> 6, c = t & 63;
  float sm = 0.f, bb = 0.f;
  if (counts[v] >= 4) {                  // MIN_POINTS
    sm = segmax[(size_t)v * 64 + c];
    int ai = min(argidx[(size_t)v * 64 + c], N - 1);
    bb = dist[ai].w;                     // bary of argmax row
  }
  out[(size_t)v * 128 + c] = sm;
  out[(size_t)v * 128 + 64 + c] = bb;
}

static inline int imin(int a, int b) { return a < b ? a : b; }

extern "C" void kernel_launch(void* const* d_in, const int* in_sizes, int n_in,
                              void* d_out, int out_size, void* d_ws, size_t ws_size,
                              hipStream_t stream) {
  const float* dist   = (const float*)d_in[0];
  const int*   indices = (const int*)d_in[1];
  const float* W0    = (const float*)d_in[3];
  const float* W1    = (const float*)d_in[4];
  const float* W2    = (const float*)d_in[5];
  const float* b2    = (const float*)d_in[6];
  const float* g0    = (const float*)d_in[7];
  const float* beta0 = (const float*)d_in[8];
  const float* g1    = (const float*)d_in[9];
  const float* beta1 = (const float*)d_in[10];

  const int N = in_sizes[0] / 4;     // distributed is N x 4
  const int V = out_size / 128;      // output is V x 128

  char* w = (char*)d_ws;
  float* segmax = (float*)w;  w += (size_t)V * 64 * sizeof(float);
  int*   argidx = (int*)w;    w += (size_t)V * 64 * sizeof(int);
  int*   counts = (int*)w;    w += ((size_t)V * sizeof(int) + 255) & ~(size_t)255;
  float* stats  = (float*)w;  w += 256 * sizeof(float);
  float* P0     = (float*)w;  w += 64 * 4 * sizeof(float);
  float* P1     = (float*)w;  w += 64 * 2 * sizeof(float);

  float* out = (float*)d_out;
  const int tot = V * 64;

  k_init<<<(tot + 255) / 256, 256, 0, stream>>>(segmax, argidx, counts, stats, V, N);
  k_pass1<<<imin((N + 255) / 256, 2048), 256, 0, stream>>>(
      (const float4*)dist, indices, counts, stats, N, V);
  k_params0<<<1, 64, 0, stream>>>(stats, W0, g0, beta0, P0, N);
  k_stats1<<<2048, TILE_THREADS, 0, stream>>>((const float4*)dist, P0, W1, stats, N);
  k_params1<<<1, 64, 0, stream>>>(stats, g1, beta1, P1, N);
  k_pass3<<<2048, TILE_THREADS, 0, stream>>>((const float4*)dist, indices, P0, W1,
                                             W2, P1, b2, segmax, argidx, N, V, 0);
  k_pass3<<<2048, TILE_THREADS, 0, stream>>>((const float4*)dist, indices, P0, W1,
                                             W2, P1, b2, segmax, argidx, N, V, 1);
  k_finalize<<<(tot + 255) / 256, 256, 0, stream>>>((const float4*)dist, segmax,
                                                    argidx, counts, out, V, N);
}